// MyGNN_31233002176552
// MI455X (gfx1250) — compile-verified
//
#include <hip/hip_runtime.h>
#include <hip/hip_bf16.h>

#define NNODES   10000
#define NEDGES   640000
#define ETOT     (NEDGES + NNODES)   // reference appends self-loops
#define FEAT     128
#define NCLS     2
#define NEG_SLOPE 0.2f

typedef __attribute__((ext_vector_type(2))) float v2f;
typedef __attribute__((ext_vector_type(8))) float v8f;

// ---------------- WMMA fp32 GEMM: C[M,N] = A[M,128] @ BT[N,128]^T -------------
// One wave32 per 16x16 output tile. K=128 fully unrolled -> 32 chained
// V_WMMA_F32_16X16X4_F32 per wave. B is pre-transposed (column-major), so both
// A and B K-pairs load as single global_load_b64 with immediate offsets.
// Requires: grid covers exactly M/16 x Ntiles; BT rows >= tn+16.
template <int LDC, int NV>
__global__ __launch_bounds__(32)
void wmma_gemm_f32(const float* __restrict__ A, const float* __restrict__ BT,
                   float* __restrict__ C) {
  const int tm   = blockIdx.x << 4;
  const int tn   = blockIdx.y << 4;
  const int lane = threadIdx.x;      // 0..31, EXEC all-ones (WMMA requirement)
  const int half = lane >> 4;        // half-wave selects K pair
  const int lid  = lane & 15;
  const int arow = tm + lid;         // A: lane holds one M row
  const int bcol = tn + lid;         // B: lane holds one N column
  const float* Ap = A  + arow * FEAT + half * 2;  // K = FEAT = 128
  const float* Bp = BT + bcol * FEAT + half * 2;  // transposed: contiguous K
  v8f acc = {};
  #pragma unroll
  for (int k0 = 0; k0 < FEAT; k0 += 4) {
    v2f a = *(const v2f*)(Ap + k0);               // one b64 load (K pair)
    v2f b = *(const v2f*)(Bp + k0);               // one b64 load (K pair)
    acc = __builtin_amdgcn_wmma_f32_16x16x4_f32(false, a, false, b,
                                                (short)0, acc, false, false);
  }
  // C/D layout: VGPR r -> M = r (+8 for upper half-wave), N = lid
  if (bcol < NV) {
    float* Cp = C + (tm + half * 8) * LDC + bcol;
    #pragma unroll
    for (int r = 0; r < 8; ++r) Cp[r * LDC] = acc[r];
  }
}

// transpose W1 [128,128] -> W1T [128,128] (one-time, L2-resident reuse)
__global__ void transpose_w1(const float* __restrict__ W, float* __restrict__ WT) {
  const int i = blockIdx.x * blockDim.x + threadIdx.x;   // FEAT*FEAT
  if (i >= FEAT * FEAT) return;
  const int k = i >> 7, n = i & 127;
  WT[n * FEAT + k] = W[k * FEAT + n];
}

// zero-pad + transpose W2 [128,2] into W2T_pad [16,128]
__global__ void pad_w2t(const float* __restrict__ W2, float* __restrict__ Wp) {
  const int i = blockIdx.x * blockDim.x + threadIdx.x;   // 16*FEAT
  if (i >= 16 * FEAT) return;
  const int n = i >> 7, k = i & 127;
  Wp[i] = (n < NCLS) ? W2[k * NCLS + n] : 0.0f;
}

// ---------------- per-node attention logits: a_s[n]=h[n].as, a_d[n]=h[n].ad
__global__ void alpha_kernel(const float* __restrict__ h,
                             const float* __restrict__ avs,
                             const float* __restrict__ avd,
                             float* __restrict__ as_out,
                             float* __restrict__ ad_out,
                             int n_nodes, int feat) {
  const int gid  = blockIdx.x * blockDim.x + threadIdx.x;
  const int node = gid >> 5;
  const int lane = gid & 31;
  if (node >= n_nodes) return;
  float s = 0.0f, d = 0.0f;
  for (int f = lane; f < feat; f += 32) {
    const float hv = h[(size_t)node * feat + f];
    s += hv * avs[f];
    d += hv * avd[f];
  }
  #pragma unroll
  for (int off = 16; off; off >>= 1) {
    s += __shfl_xor(s, off, 32);
    d += __shfl_xor(d, off, 32);
  }
  if (lane == 0) { as_out[node] = s; ad_out[node] = d; }
}

// ---------------- helpers ----------------
__device__ __forceinline__ int f2key(float f) {        // monotone float->int
  int i = __float_as_int(f);
  return (i >= 0) ? i : (i ^ 0x7FFFFFFF);
}
__device__ __forceinline__ float key2f(int k) {
  int i = (k >= 0) ? k : (k ^ 0x7FFFFFFF);
  return __int_as_float(i);
}
__device__ __forceinline__ void edge_sd(const int* __restrict__ ei, int e,
                                        int& s, int& d) {
  if (e < NEDGES) { s = ei[e]; d = ei[NEDGES + e]; }
  else            { s = d = e - NEDGES; }              // self loop
}
__device__ __forceinline__ float lrelu(float v) {
  return v > 0.0f ? v : NEG_SLOPE * v;
}

__global__ void init_seg(int* __restrict__ mkey, float* __restrict__ denom, int n) {
  const int i = blockIdx.x * blockDim.x + threadIdx.x;
  if (i < n) { mkey[i] = f2key(-__builtin_huge_valf()); denom[i] = 0.0f; }
}
__global__ void zero_f32(float* __restrict__ p, int n) {
  const int i = blockIdx.x * blockDim.x + threadIdx.x;
  if (i < n) p[i] = 0.0f;
}

// ---------------- edge pass 1: segment max of leaky-relu logits ----------------
__global__ void edge_max(const int* __restrict__ ei, const float* __restrict__ as,
                         const float* __restrict__ ad, int* __restrict__ mkey) {
  const int e = blockIdx.x * blockDim.x + threadIdx.x;
  if (e >= ETOT) return;
  int s, d; edge_sd(ei, e, s, d);
  atomicMax(&mkey[d], f2key(lrelu(as[s] + ad[d])));
}

// ---------------- edge pass 2: segment sum of exp(e - m[dst]) ----------------
__global__ void edge_denom(const int* __restrict__ ei, const float* __restrict__ as,
                           const float* __restrict__ ad, const int* __restrict__ mkey,
                           float* __restrict__ denom) {
  const int e = blockIdx.x * blockDim.x + threadIdx.x;
  if (e >= ETOT) return;
  int s, d; edge_sd(ei, e, s, d);
  const float v = lrelu(as[s] + ad[d]);
  atomicAdd(&denom[d], __expf(v - key2f(mkey[d])));
}

// ---------------- edge pass 3 (layer 1): wave32/edge, 128-feature scatter ----
__global__ void edge_agg128(const int* __restrict__ ei, const float* __restrict__ as,
                            const float* __restrict__ ad, const int* __restrict__ mkey,
                            const float* __restrict__ denom,
                            const float* __restrict__ h, float* __restrict__ out) {
  const int gid  = blockIdx.x * blockDim.x + threadIdx.x;
  const int e    = gid >> 5;
  const int lane = gid & 31;
  if (e >= ETOT) return;
  int s, d; edge_sd(ei, e, s, d);
  const float v = lrelu(as[s] + ad[d]);
  const float w = __expf(v - key2f(mkey[d])) / denom[d];
  const float* hs = h   + (size_t)s * FEAT;
  float*       od = out + (size_t)d * FEAT;
  #pragma unroll
  for (int j = 0; j < FEAT / 32; ++j) {
    const int f = lane + 32 * j;
    atomicAdd(&od[f], hs[f] * w);
  }
}

// ---------------- edge pass 3 (layer 2): thread/edge, 2 features ----------------
__global__ void edge_agg2(const int* __restrict__ ei, const float* __restrict__ as,
                          const float* __restrict__ ad, const int* __restrict__ mkey,
                          const float* __restrict__ denom,
                          const float* __restrict__ z, float* __restrict__ out) {
  const int e = blockIdx.x * blockDim.x + threadIdx.x;
  if (e >= ETOT) return;
  int s, d; edge_sd(ei, e, s, d);
  const float v = lrelu(as[s] + ad[d]);
  const float w = __expf(v - key2f(mkey[d])) / denom[d];
  atomicAdd(&out[(size_t)d * NCLS + 0], z[(size_t)s * NCLS + 0] * w);
  atomicAdd(&out[(size_t)d * NCLS + 1], z[(size_t)s * NCLS + 1] * w);
}

// ---------------- bias (+optional ReLU) ----------------
__global__ void bias_act(const float* __restrict__ in, const float* __restrict__ bias,
                         float* __restrict__ out, int total, int feat, int do_relu) {
  const int i = blockIdx.x * blockDim.x + threadIdx.x;
  if (i >= total) return;
  const float v = in[i] + bias[i % feat];
  out[i] = do_relu ? fmaxf(v, 0.0f) : v;
}

extern "C" void kernel_launch(void* const* d_in, const int* in_sizes, int n_in,
                              void* d_out, int out_size, void* d_ws, size_t ws_size,
                              hipStream_t stream) {
  (void)in_sizes; (void)n_in; (void)out_size; (void)ws_size;
  const float* x    = (const float*)d_in[0];
  const int*   ei   = (const int*)  d_in[1];   // [2, NEDGES]: src row then dst row
  const float* W1   = (const float*)d_in[2];
  const float* asv1 = (const float*)d_in[3];
  const float* adv1 = (const float*)d_in[4];
  const float* b1   = (const float*)d_in[5];
  const float* W2   = (const float*)d_in[6];
  const float* asv2 = (const float*)d_in[7];
  const float* adv2 = (const float*)d_in[8];
  const float* b2   = (const float*)d_in[9];
  float* out = (float*)d_out;

  // workspace layout (floats); total ~2.68M floats = 10.7 MB
  float* ws    = (float*)d_ws;
  float* h1    = ws;                         // [NNODES*FEAT]
  float* agg1  = h1   + (size_t)NNODES*FEAT; // [NNODES*FEAT] (becomes h2)
  float* as1   = agg1 + (size_t)NNODES*FEAT; // [NNODES]
  float* ad1   = as1  + NNODES;
  int*   mkey  = (int*)(ad1 + NNODES);       // [NNODES]
  float* denom = (float*)(mkey + NNODES);    // [NNODES]
  float* z2    = denom + NNODES;             // [NNODES*NCLS]
  float* as2   = z2   + (size_t)NNODES*NCLS;
  float* ad2   = as2  + NNODES;
  float* agg2  = ad2  + NNODES;              // [NNODES*NCLS]
  float* w1t   = agg2 + (size_t)NNODES*NCLS; // [FEAT*FEAT] W1 transposed
  float* w2tp  = w1t  + (size_t)FEAT*FEAT;   // [16*FEAT] W2 transposed+padded

  const int B = 256;
  const int nodeB  = (NNODES + B - 1) / B;
  const int edgeB  = (ETOT + B - 1) / B;
  const int waveNB = (NNODES * 32 + B - 1) / B;
  const int waveEB = (ETOT * 32 + B - 1) / B;
  const int featB  = (NNODES * FEAT + B - 1) / B;
  const int clsB   = (NNODES * NCLS + B - 1) / B;

  // ===== Layer 1 =====
  transpose_w1<<<(FEAT * FEAT + B - 1) / B, B, 0, stream>>>(W1, w1t);
  wmma_gemm_f32<FEAT, FEAT>
      <<<dim3(NNODES / 16, FEAT / 16), 32, 0, stream>>>(x, w1t, h1);
  alpha_kernel<<<waveNB, B, 0, stream>>>(h1, asv1, adv1, as1, ad1, NNODES, FEAT);
  init_seg<<<nodeB, B, 0, stream>>>(mkey, denom, NNODES);
  zero_f32<<<featB, B, 0, stream>>>(agg1, NNODES * FEAT);
  edge_max  <<<edgeB, B, 0, stream>>>(ei, as1, ad1, mkey);
  edge_denom<<<edgeB, B, 0, stream>>>(ei, as1, ad1, mkey, denom);
  edge_agg128<<<waveEB, B, 0, stream>>>(ei, as1, ad1, mkey, denom, h1, agg1);
  bias_act<<<featB, B, 0, stream>>>(agg1, b1, agg1, NNODES * FEAT, FEAT, 1); // +b1, ReLU

  // ===== Layer 2 =====
  pad_w2t<<<(16 * FEAT + B - 1) / B, B, 0, stream>>>(W2, w2tp);
  wmma_gemm_f32<NCLS, NCLS>
      <<<dim3(NNODES / 16, 1), 32, 0, stream>>>(agg1, w2tp, z2);
  alpha_kernel<<<waveNB, B, 0, stream>>>(z2, asv2, adv2, as2, ad2, NNODES, NCLS);
  init_seg<<<nodeB, B, 0, stream>>>(mkey, denom, NNODES);
  zero_f32<<<clsB, B, 0, stream>>>(agg2, NNODES * NCLS);
  edge_max  <<<edgeB, B, 0, stream>>>(ei, as2, ad2, mkey);
  edge_denom<<<edgeB, B, 0, stream>>>(ei, as2, ad2, mkey, denom);
  edge_agg2 <<<edgeB, B, 0, stream>>>(ei, as2, ad2, mkey, denom, z2, agg2);
  bias_act<<<clsB, B, 0, stream>>>(agg2, b2, out, NNODES * NCLS, NCLS, 0);   // +b2
}